// NGCFModel_14766097564250
// MI455X (gfx1250) — compile-verified
//
#include <hip/hip_runtime.h>
#include <hip/hip_bf16.h>

typedef __attribute__((ext_vector_type(2))) float v2f;
typedef __attribute__((ext_vector_type(4))) float v4f;
typedef __attribute__((ext_vector_type(8))) float v8f;

#define U_DIM 256
#define I_DIM 1024
#define B_DIM 512
#define E_DIM 64

// Workspace byte offsets
#define OFF_RS    0u                         // 256*1024 floats = 1 MiB
#define OFF_LIST  (U_DIM * I_DIM * 4u)       // 512 ints (packed w<<16|row)
#define OFF_NROWS (OFF_LIST + B_DIM * 4u)    // 1 int
#define OFF_USED  (OFF_NROWS + 4u)           // 256 ints

// ---------------------------------------------------------------------------
// Kernel 0: items histogram -> deterministic compacted (weight,row) list via
// Hillis-Steele prefix scan (ascending row order => deterministic FP sums),
// plus used-user flags. Single block, 1024 threads.
// ---------------------------------------------------------------------------
__global__ __launch_bounds__(1024) void ngcf_prep_kernel(const int* __restrict__ users,
                                                         const int* __restrict__ items,
                                                         int* __restrict__ list,
                                                         int* __restrict__ nrows,
                                                         int* __restrict__ used) {
    __shared__ int s_cnt[I_DIM];
    __shared__ int s_scan[I_DIM];
    __shared__ int s_used[U_DIM];
    const int tid = threadIdx.x;

    s_cnt[tid] = 0;
    if (tid < U_DIM) s_used[tid] = 0;
    __syncthreads();
    if (tid < B_DIM) {
        atomicAdd(&s_cnt[items[tid] & (I_DIM - 1)], 1);
        s_used[users[tid] & (U_DIM - 1)] = 1;   // all writers store 1 -> race-free
    }
    __syncthreads();

    const int w    = s_cnt[tid];
    const int flag = (w != 0) ? 1 : 0;
    s_scan[tid] = flag;
    __syncthreads();
    // Inclusive Hillis-Steele scan over 1024 flags
    for (int off = 1; off < I_DIM; off <<= 1) {
        const int v = s_scan[tid];
        const int a = (tid >= off) ? s_scan[tid - off] : 0;
        __syncthreads();
        s_scan[tid] = v + a;
        __syncthreads();
    }
    if (flag) list[s_scan[tid] - 1] = (w << 16) | tid;   // rows ascending
    if (tid == 0) nrows[0] = s_scan[I_DIM - 1];
    if (tid < U_DIM) used[tid] = s_used[tid];
}

// ---------------------------------------------------------------------------
// Kernel 1: RS[u,:] = sum over compacted list of w * T[u,row,:]
// The HBM-roofline kernel (~400 rows x 4KB x ~221 users ~ 360 MB @ 23.3 TB/s).
// Dense branch-free loop, unrolled x4, non-temporal streaming loads.
// ---------------------------------------------------------------------------
__global__ __launch_bounds__(256) void ngcf_rowsum_kernel(const float* __restrict__ T,
                                                          const int* __restrict__ list,
                                                          const int* __restrict__ nrows_p,
                                                          const int* __restrict__ used,
                                                          float* __restrict__ RS) {
    const int u = blockIdx.x;
    if (!used[u]) return;                       // uniform across block

    __shared__ int s_list[B_DIM];
    const int tid = threadIdx.x;
    const int n   = nrows_p[0];
    for (int i = tid; i < n; i += 256) s_list[i] = list[i];
    __syncthreads();

    const float* __restrict__ Tu = T + (size_t)u * I_DIM * I_DIM;
    const int col = tid * 4;
    v4f acc = {0.f, 0.f, 0.f, 0.f};

#pragma unroll 4
    for (int i = 0; i < n; ++i) {
        const int   e = s_list[i];
        const float w = (float)(e >> 16);
        const v4f   v = __builtin_nontemporal_load(
            (const v4f*)(Tu + (size_t)(e & 0xffff) * I_DIM + col));
        acc += w * v;
    }
    *(v4f*)(RS + (size_t)u * I_DIM + col) = acc;
}

// ---------------------------------------------------------------------------
// Kernel 2: sel = RS[users,:][:,items];  gamma = sel @ gi  (WMMA f32 16x16x4);
// row-normalize; xui = rowdot(gu, gamma_n); emit (xui | gu | gamma_n).
// Grid: 32 blocks (16 b-rows each) x 128 threads (4 waves; wave w -> N cols
// 16w..16w+15).  K loop: 512 in steps of 4 -> 128 v_wmma per wave.
// ---------------------------------------------------------------------------
__global__ __launch_bounds__(128) void ngcf_gemm_norm_kernel(const float* __restrict__ RS,
                                                             const float* __restrict__ gu,
                                                             const float* __restrict__ gi,
                                                             const int* __restrict__ users,
                                                             const int* __restrict__ items,
                                                             float* __restrict__ out) {
    __shared__ int   s_items[B_DIM];
    __shared__ float s_g[16][E_DIM];
    __shared__ float s_norm[16];
    __shared__ float s_part[128];

    float* __restrict__ out_xui = out;                       // [512]
    float* __restrict__ out_gu  = out + B_DIM;               // [512*64]
    float* __restrict__ out_gii = out + B_DIM + B_DIM*E_DIM; // [512*64]

    const int tid   = threadIdx.x;
    const int lane  = tid & 31;
    const int wave  = tid >> 5;          // 0..3
    const int m0    = blockIdx.x * 16;   // b-row base
    const int n0    = wave * 16;         // e-col base
    const int khalf = lane >> 4;         // 0|1 : K sub-pair select
    const int l15   = lane & 15;

    for (int i = tid; i < B_DIM; i += 128) s_items[i] = items[i];
    __syncthreads();

    // A-fragment source row: lane l (and l+16) owns M = m0 + (l&15)
    const int    urow  = users[m0 + l15] & (U_DIM - 1);
    const float* rsrow = RS + (size_t)urow * I_DIM;
    const int    ncol  = n0 + l15;       // B-fragment / D column

    v8f acc = {0.f, 0.f, 0.f, 0.f, 0.f, 0.f, 0.f, 0.f};

    for (int k0 = 0; k0 < B_DIM; k0 += 4) {
        const int ka = k0 + 2 * khalf;
        v2f a, b;
        a.x = rsrow[s_items[ka]];        // A[m, ka]   = sel on the fly
        a.y = rsrow[s_items[ka + 1]];    // A[m, ka+1]
        b.x = gi[(size_t)ka * E_DIM + ncol];        // B[ka,   n]
        b.y = gi[(size_t)(ka + 1) * E_DIM + ncol];  // B[ka+1, n]
        acc = __builtin_amdgcn_wmma_f32_16x16x4_f32(
            /*neg_a=*/false, a, /*neg_b=*/false, b,
            /*c_mod=*/(short)0, acc, /*reuse_a=*/false, /*reuse_b=*/false);
    }

    // D layout: VGPR r -> local row khalf*8 + r, local col = lane&15
    const int rbase = khalf * 8;
#pragma unroll
    for (int r = 0; r < 8; ++r) {
        s_g[rbase + r][n0 + l15] = acc[r];
    }
    __syncthreads();

    // Row L2 norms over E=64 (16 rows, threads 0..15)
    if (tid < 16) {
        float ss = 0.f;
        for (int c = 0; c < E_DIM; ++c) {
            const float g = s_g[tid][c];
            ss = fmaf(g, g, ss);
        }
        s_norm[tid] = fmaxf(sqrtf(ss), 1e-12f);
    }
    __syncthreads();

    // Each thread: 8 contiguous cols of one row. Normalize, emit, partial dot.
    const int row = tid >> 3;            // 0..15
    const int c0  = (tid & 7) * 8;       // 0..56
    const int gm  = m0 + row;
    const float inv = 1.0f / s_norm[row];
    float part = 0.f;
#pragma unroll
    for (int i = 0; i < 8; ++i) {
        const int c   = c0 + i;
        const float gn  = s_g[row][c] * inv;
        const float guv = gu[(size_t)gm * E_DIM + c];
        out_gii[(size_t)gm * E_DIM + c] = gn;
        out_gu [(size_t)gm * E_DIM + c] = guv;
        part = fmaf(guv, gn, part);
    }
    s_part[tid] = part;
    __syncthreads();
    if (tid < 16) {
        float s = 0.f;
#pragma unroll
        for (int i = 0; i < 8; ++i) s += s_part[tid * 8 + i];
        out_xui[m0 + tid] = s;
    }
}

// ---------------------------------------------------------------------------
extern "C" void kernel_launch(void* const* d_in, const int* in_sizes, int n_in,
                              void* d_out, int out_size, void* d_ws, size_t ws_size,
                              hipStream_t stream) {
    const float* T     = (const float*)d_in[0];   // (256,1024,1024)
    const float* gu    = (const float*)d_in[1];   // (512,64)
    const float* gi    = (const float*)d_in[2];   // (512,64)
    const int*   users = (const int*)d_in[3];     // (512,)
    const int*   items = (const int*)d_in[4];     // (512,)
    float*       out   = (float*)d_out;           // 512 + 32768 + 32768

    char* ws = (char*)d_ws;
    float* RS    = (float*)(ws + OFF_RS);
    int*   list  = (int*)(ws + OFF_LIST);
    int*   nrows = (int*)(ws + OFF_NROWS);
    int*   used  = (int*)(ws + OFF_USED);

    ngcf_prep_kernel<<<1, 1024, 0, stream>>>(users, items, list, nrows, used);
    ngcf_rowsum_kernel<<<U_DIM, 256, 0, stream>>>(T, list, nrows, used, RS);
    ngcf_gemm_norm_kernel<<<B_DIM / 16, 128, 0, stream>>>(RS, gu, gi, users, items, out);
}